// MultiLabelCrossEntropyLabelSmooth_82446192214245
// MI455X (gfx1250) — compile-verified
//
#include <hip/hip_runtime.h>
#include <hip/hip_bf16.h>
#include <math.h>

#define NPAR   16
#define NTOTAL 5776
#define NBATCH 8192
#define LS_EPS 0.1f

// children counts / start offsets per parent class (from reference)
__constant__ int g_counts[16] = {128,256,512,384,192,640,320,448,256,512,128,384,576,320,448,256};
__constant__ int g_starts[16] = {0,128,384,896,1280,1472,2112,2432,2880,3136,3648,3776,4160,4736,5056,5504};

typedef __attribute__((ext_vector_type(2))) float v2f;
typedef __attribute__((ext_vector_type(8))) float v8f;

// One wave32 per sample: single-pass online softmax over the 16 parent logits and
// the sample's child range. All child counts are multiples of 64 -> float2 loads,
// no tail masking, every lane does cnt/64 uniform iterations.
__global__ __launch_bounds__(256) void hier_ce_partial(
    const float* __restrict__ predicts,
    const int*   __restrict__ parent_t,
    const int*   __restrict__ child_t,
    float*       __restrict__ partial)      // [gridDim.x] per-block sums
{
    __shared__ float wsum[8];
    const int lane = threadIdx.x & 31;
    const int wid  = threadIdx.x >> 5;
    const int wpb  = blockDim.x >> 5;
    const int gw   = blockIdx.x * wpb + wid;
    const int nw   = gridDim.x * wpb;

    float acc = 0.0f;
    for (int b = gw; b < NBATCH; b += nw) {
        const float* __restrict__ row = predicts + (size_t)b * NTOTAL;
        const int pt  = parent_t[b];
        const int ct  = child_t[b];
        const int cnt = g_counts[pt];
        const int st  = g_starts[pt];

        if (b + nw < NBATCH)  // warm next sample's row (global_prefetch_b8)
            __builtin_prefetch(predicts + (size_t)(b + nw) * NTOTAL, 0, 1);

        // ---- parent softmax (16 wide): lanes 16..31 carry neutral values ----
        float xp  = (lane < NPAR) ? row[lane] : -INFINITY;
        float mp  = xp;
        float sp  = (lane < NPAR) ? 1.0f : 0.0f;
        float sxp = (lane < NPAR) ? xp   : 0.0f;
        float xtp = (lane == pt)  ? xp   : 0.0f;

        // ---- child range: float2 online max / sum-exp / sum / target pick ----
        const float2* __restrict__ crow2 =
            (const float2*)(row + NPAR + st);           // 16B-aligned (st % 64 == 0)
        const int npair = cnt >> 1;                      // multiple of 32
        float mc = -INFINITY, sc = 0.0f, syc = 0.0f, ytc = 0.0f;
        for (int p = lane; p < npair; p += 32) {
            float2 y = crow2[p];
            syc += y.x + y.y;
            const int i0 = p << 1;
            if (i0     == ct) ytc = y.x;
            if (i0 + 1 == ct) ytc = y.y;
            float mn = fmaxf(mc, fmaxf(y.x, y.y));       // one max-merge per pair
            sc = sc * __expf(mc - mn) + __expf(y.x - mn) + __expf(y.y - mn);
            mc = mn;
        }

        // ---- wave32 butterfly reduction (first step pairs lane L with L^16,
        //      so -inf neutrals always meet a finite partner: no NaN) ----
        #pragma unroll
        for (int o = 16; o > 0; o >>= 1) {
            float mo = __shfl_xor(mp, o);
            float so = __shfl_xor(sp, o);
            float mn = fmaxf(mp, mo);
            sp = sp * __expf(mp - mn) + so * __expf(mo - mn);
            mp = mn;
            sxp += __shfl_xor(sxp, o);
            xtp += __shfl_xor(xtp, o);

            float mo2 = __shfl_xor(mc, o);
            float so2 = __shfl_xor(sc, o);
            float mn2 = fmaxf(mc, mo2);
            sc = sc * __expf(mc - mn2) + so2 * __expf(mo2 - mn2);
            mc = mn2;
            syc += __shfl_xor(syc, o);
            ytc += __shfl_xor(ytc, o);
        }

        if (lane == 0) {
            // L = (m + logZ) - (1-eps)*x[t] - (eps/c)*sum(x)
            float lp = (mp + __logf(sp)) - (1.0f - LS_EPS) * xtp - (LS_EPS / 16.0f) * sxp;
            float lc = (mc + __logf(sc)) - (1.0f - LS_EPS) * ytc - (LS_EPS / (float)cnt) * syc;
            acc += lp + lc;
        }
    }

    if (lane == 0) wsum[wid] = acc;
    __syncthreads();
    if (threadIdx.x == 0) {
        float t = 0.0f;
        for (int w = 0; w < wpb; ++w) t += wsum[w];
        partial[blockIdx.x] = t;
    }
}

// Single wave32 (EXEC all-1s as WMMA requires): grid-stride partial sums into a
// 16x4 f32 A-matrix, multiply by all-ones B via v_wmma_f32_16x16x4_f32; D row-sums
// collapse the 64 slots. Deterministic (fixed order, no atomics).
__global__ __launch_bounds__(32) void hier_ce_finalize(
    const float* __restrict__ partial, int n, float* __restrict__ out)
{
    const int lane = threadIdx.x;
    float a0 = 0.0f, a1 = 0.0f;
    for (int i = lane;      i < n; i += 64) a0 += partial[i];
    for (int i = lane + 32; i < n; i += 64) a1 += partial[i];

    v2f av; av.x = a0;   av.y = a1;
    v2f bv; bv.x = 1.0f; bv.y = 1.0f;   // ones B: layout-independent row-sum trick
    v8f c = {};
    c = __builtin_amdgcn_wmma_f32_16x16x4_f32(
            /*neg_a=*/false, av, /*neg_b=*/false, bv,
            /*c_mod=*/(short)0, c, /*reuse_a=*/false, /*reuse_b=*/false);

    float rs = c[0] + c[1] + c[2] + c[3] + c[4] + c[5] + c[6] + c[7]; // rows M=0..7 (lane<16) / M=8..15 (lane>=16) at N=lane%16
    float rs_hi = __shfl(rs, 16, 32);
    if (lane == 0) out[0] = (rs + rs_hi) * (1.0f / (float)NBATCH);
}

extern "C" void kernel_launch(void* const* d_in, const int* in_sizes, int n_in,
                              void* d_out, int out_size, void* d_ws, size_t ws_size,
                              hipStream_t stream) {
    const float* predicts = (const float*)d_in[0];
    const int*   parent_t = (const int*)d_in[1];
    const int*   child_t  = (const int*)d_in[2];
    float*       out      = (float*)d_out;
    float*       partial  = (float*)d_ws;

    int nb = 1024;                               // 1024 blocks x 8 waves = 8192 waves (1 sample/wave)
    size_t cap = ws_size / sizeof(float);
    if ((size_t)nb > cap) nb = (int)cap;
    if (nb < 1) nb = 1;

    hier_ce_partial<<<nb, 256, 0, stream>>>(predicts, parent_t, child_t, partial);
    hier_ce_finalize<<<1, 32, 0, stream>>>(partial, nb, out);
}